// SkipInteractionBlock_13503377179219
// MI455X (gfx1250) — compile-verified
//
#include <hip/hip_runtime.h>

// SkipInteractionBlock for MI455X (gfx1250), wave32 + V_WMMA_F32_16X16X4_F32.
//
// Decomposition:
//  K0  W_all[k, l*64+j] = a_tp^2 * (W_conv[l] @ L1[l])          (tiny)
//  Kz  zero the m scatter accumulator (atomics need zeros per call)
//  K1  sw_all = node_feats @ W_all            25000x64 @ 64x192   (WMMA f32)
//  K2  m[recv] += sw_all[send] (x) Y          edge scatter, global f32 atomics
//  K3  per (node,channel)-row tile of 16:
//        x_mid = (m rows) @ W_skip_l, contracted with a_skip*attrs (WMMA f32)
//        x2    = x_mid @ L2_l                                     (WMMA f32)
//        out   = m + a_tp * x2   (scattered to k*d_l+m layout)
//      Both GEMMs fused through LDS -> no x_mid HBM round trip.

#define NNODES 25000
#define NEDGES 400000
#define FEATC  576            // 64 * (1+3+5)
#define A_TP   0.125f
#define A_TP2  0.015625f
#define A_SKIP 0.03952847075210474f   // 1/sqrt(64*10)

typedef __attribute__((ext_vector_type(2))) float v2f;
typedef __attribute__((ext_vector_type(8))) float v8f;

__device__ __forceinline__ v8f wmma4(v2f a, v2f b, v8f c) {
  // D = A(16x4) * B(4x16) + C(16x16), fp32
  return __builtin_amdgcn_wmma_f32_16x16x4_f32(false, a, false, b, (short)0, c,
                                               false, false);
}

// ---------------- K0: W_all = a_tp^2 * W_conv @ L1, packed 64 x 192 ---------
__global__ __launch_bounds__(256) void k_wall(const float* __restrict__ Wc,
                                              const float* __restrict__ L1,
                                              float* __restrict__ Wall) {
  const int l = blockIdx.x;
  const float* A = Wc + l * 4096;
  const float* B = L1 + l * 4096;
  for (int idx = threadIdx.x; idx < 4096; idx += blockDim.x) {
    const int k = idx >> 6, j = idx & 63;
    float s = 0.f;
    #pragma unroll 8
    for (int t = 0; t < 64; ++t) s += A[k * 64 + t] * B[t * 64 + j];
    Wall[k * 192 + l * 64 + j] = A_TP2 * s;
  }
}

// ---------------- Kz: zero fill ---------------------------------------------
__global__ __launch_bounds__(256) void k_zero(float* __restrict__ p, long n) {
  long i = (long)blockIdx.x * blockDim.x + threadIdx.x;
  const long stride = (long)gridDim.x * blockDim.x;
  for (; i < n; i += stride) p[i] = 0.f;
}

// ---------------- K1: sw_all = node_feats @ W_all (WMMA) --------------------
__global__ __launch_bounds__(32) void k_sw(const float* __restrict__ nf,
                                           const float* __restrict__ Wall,
                                           float* __restrict__ sw) {
  const int lane = threadIdx.x & 31;
  const int half = lane >> 4, lr = lane & 15;
  const int rt = blockIdx.x;   // row tile (16 nodes)
  const int ct = blockIdx.y;   // col tile (16 of 192)
  const int arow = min(rt * 16 + lr, NNODES - 1);
  const float* ap = nf + (long)arow * 64;
  v8f acc = {};
  #pragma unroll
  for (int ks = 0; ks < 16; ++ks) {
    const int kb = ks * 4 + half * 2;
    v2f a, b;
    a.x = ap[kb];
    a.y = ap[kb + 1];
    b.x = Wall[kb * 192 + ct * 16 + lr];
    b.y = Wall[(kb + 1) * 192 + ct * 16 + lr];
    acc = wmma4(a, b, acc);
  }
  #pragma unroll
  for (int e = 0; e < 8; ++e) {
    const int row = rt * 16 + e + half * 8;
    if (row < NNODES) sw[(long)row * 192 + ct * 16 + lr] = acc[e];
  }
}

// ---------------- K2: edge scatter with global f32 atomics ------------------
__global__ __launch_bounds__(256) void k_edge(const int* __restrict__ eidx,
                                              const float* __restrict__ ef,
                                              const float* __restrict__ sw,
                                              float* __restrict__ m) {
  const int wave = threadIdx.x >> 5;
  const int lane = threadIdx.x & 31;
  const long e = (long)blockIdx.x * 8 + wave;
  if (e >= NEDGES) return;
  const long s = (long)eidx[e];
  const long r = (long)eidx[NEDGES + e];
  const float* swp = sw + s * 192;
  float s0[3], s1[3];
  #pragma unroll
  for (int l = 0; l < 3; ++l) {
    s0[l] = swp[l * 64 + lane];
    s1[l] = swp[l * 64 + lane + 32];
  }
  float* mp = m + r * FEATC;
  const float* yp = ef + e * 9;
  #pragma unroll
  for (int c = 0; c < 9; ++c) {
    const int l = (c == 0) ? 0 : ((c < 4) ? 1 : 2);
    const float y = yp[c];
    unsafeAtomicAdd(mp + c * 64 + lane, s0[l] * y);
    unsafeAtomicAdd(mp + c * 64 + lane + 32, s1[l] * y);
  }
}

// ---------------- K3: fused skip path (two WMMA GEMMs through LDS) ----------
__global__ __launch_bounds__(32) void k_node(const float* __restrict__ m,
                                             const float* __restrict__ attrs,
                                             const float* __restrict__ Wskip,
                                             const float* __restrict__ L2,
                                             float* __restrict__ out) {
  __shared__ float sm[16][68];   // 16 (n,c)-rows of m, K=64 (padded stride)
  __shared__ float xm[16][68];   // x_mid tile
  __shared__ float sat[16][12];  // a_skip * attrs per row

  const int l = blockIdx.y;
  const int d = (l == 0) ? 1 : ((l == 1) ? 3 : 5);
  const int c0 = (l == 0) ? 0 : ((l == 1) ? 1 : 4);
  const int off = (l == 0) ? 0 : ((l == 1) ? 64 : 256);
  const int nrows = NNODES * d;
  const int rt = blockIdx.x;
  if (rt * 16 >= nrows) return;

  const int lane = threadIdx.x & 31;
  const int half = lane >> 4, lr = lane & 15;

  // stage 16 m-rows (row r -> node n = r/d, channel c = c0 + r%d)
  for (int t = 0; t < 32; ++t) {
    const int idx = t * 32 + lane;
    const int rl = idx >> 6, k = idx & 63;
    const int r = min(rt * 16 + rl, nrows - 1);
    const int n = r / d, c = c0 + r % d;
    sm[rl][k] = m[(long)n * FEATC + c * 64 + k];
  }
  // stage scaled attrs per row
  for (int t = 0; t < 5; ++t) {
    const int idx = t * 32 + lane;
    const int rl = idx / 10, a = idx % 10;
    const int r = min(rt * 16 + rl, nrows - 1);
    const int n = r / d;
    sat[rl][a] = A_SKIP * attrs[n * 10 + a];
  }
  __syncthreads();

  // phase 1: x_mid[row, j] = sum_a sat[row,a] * (m_row @ Wskip[:, a*64+j])
  const float* Wsk = Wskip + l * 40960;  // [k][a][j], row stride 640
  for (int jt = 0; jt < 4; ++jt) {
    v8f acc = {};
    for (int a = 0; a < 10; ++a) {
      v8f cc = {};
      #pragma unroll
      for (int ks = 0; ks < 16; ++ks) {
        const int kb = ks * 4 + half * 2;
        v2f av, bv;
        av.x = sm[lr][kb];
        av.y = sm[lr][kb + 1];
        bv.x = Wsk[kb * 640 + a * 64 + jt * 16 + lr];
        bv.y = Wsk[(kb + 1) * 640 + a * 64 + jt * 16 + lr];
        cc = wmma4(av, bv, cc);
      }
      #pragma unroll
      for (int e = 0; e < 8; ++e) acc[e] += sat[e + half * 8][a] * cc[e];
    }
    #pragma unroll
    for (int e = 0; e < 8; ++e) xm[e + half * 8][jt * 16 + lr] = acc[e];
  }
  __syncthreads();

  // phase 2: x2 = x_mid @ L2_l ; out = m + a_tp * x2 (scattered layout)
  const float* L2l = L2 + l * 4096;
  for (int ct = 0; ct < 4; ++ct) {
    v8f cc = {};
    #pragma unroll
    for (int ks = 0; ks < 16; ++ks) {
      const int kb = ks * 4 + half * 2;
      v2f av, bv;
      av.x = xm[lr][kb];
      av.y = xm[lr][kb + 1];
      bv.x = L2l[kb * 64 + ct * 16 + lr];
      bv.y = L2l[(kb + 1) * 64 + ct * 16 + lr];
      cc = wmma4(av, bv, cc);
    }
    #pragma unroll
    for (int e = 0; e < 8; ++e) {
      const int rl = e + half * 8;
      const int r = rt * 16 + rl;
      if (r < nrows) {
        const int n = r / d, ci = r % d;
        const int u = ct * 16 + lr;  // output mul index
        out[(long)n * FEATC + off + u * d + ci] = sm[rl][u] + A_TP * cc[e];
      }
    }
  }
}

// ---------------- launch ----------------------------------------------------
extern "C" void kernel_launch(void* const* d_in, const int* in_sizes, int n_in,
                              void* d_out, int out_size, void* d_ws,
                              size_t ws_size, hipStream_t stream) {
  const float* nf    = (const float*)d_in[0];  // node_feats  (N,64)
  const float* attrs = (const float*)d_in[1];  // node_attrs  (N,10)
  const float* ef    = (const float*)d_in[2];  // edge_feats  (E,9)
  const float* Wc    = (const float*)d_in[3];  // W_conv      (3,64,64)
  const float* L1    = (const float*)d_in[4];  // L1          (3,64,64)
  const float* Wsk   = (const float*)d_in[5];  // W_skip      (3,64,10,64)
  const float* L2    = (const float*)d_in[6];  // L2          (3,64,64)
  const int*   eidx  = (const int*)d_in[7];    // edge_index  (2,E)
  float* out = (float*)d_out;

  float* ws   = (float*)d_ws;
  float* Wall = ws;                           // 12288 floats
  float* sw   = Wall + 12288;                 // N*192 floats
  float* mbuf = sw + (long)NNODES * 192;      // N*576 floats

  k_wall<<<3, 256, 0, stream>>>(Wc, L1, Wall);
  k_zero<<<2048, 256, 0, stream>>>(mbuf, (long)NNODES * FEATC);
  k_sw<<<dim3(1563, 12), 32, 0, stream>>>(nf, Wall, sw);
  k_edge<<<(NEDGES + 7) / 8, 256, 0, stream>>>(eidx, ef, sw, mbuf);
  k_node<<<dim3(7813, 3), 32, 0, stream>>>(mbuf, attrs, Wsk, L2, out);
}